// DeltaModulator1Bit_38783554683098
// MI455X (gfx1250) — compile-verified
//
#include <hip/hip_runtime.h>
#include <hip/hip_bf16.h>

// DeltaModulator1Bit on MI455X (gfx1250).
// Sequential scan over T=8192 with 4096 independent (b,c) chains.
//   curr = prev + step*sign(x - prev)  (sign(0)->+1),  bit = (sign<0) ? 1 : 0
//
// Data path (CDNA5-specific):
//  - TDM: tensor_load_to_lds streams 4KB tiles (32 t-rows x 32 ch x 4B) of x
//    into a 16-slot LDS ring per block, 15 tiles in flight per wave tracked by
//    TENSORcnt  ->  ~7.7 MB of HBM loads in flight across 128 blocks, matching
//    the 23.3 TB/s latency-BW product, at ~1 instruction per 32 timesteps.
//  - Compute reads tiles via ds_load_b32 (low latency, DScnt pipe).
//  - Outputs (268 MB) written with non-temporal stores so x (134 MB) stays
//    resident in the 192 MB L2 across graph replays.

#define DM_B    16
#define DM_T    8192
#define DM_C    256
#define DM_CPB  32                  // channels per block == blockDim.x (1 wave32)
#define CH_T    32                  // t-rows per TDM tile (tile = 4 KB)
#define NBUF    16                  // LDS ring slots (64 KB static LDS)
#define AHEAD   15                  // TDM loads in flight (< NBUF, <= 63)
#define NCHUNK  (DM_T / CH_T)       // 256 tiles per chain-block

typedef unsigned int v4u __attribute__((ext_vector_type(4)));
typedef int          v8i __attribute__((ext_vector_type(8)));
typedef int          v4i __attribute__((ext_vector_type(4)));

// Issue one TDM load: 2-D tile, 32 rows x 32 elems, element = 4 B,
// row stride = C = 256 elems. Tensor dims set huge so the in-bounds tile
// never trips the OOB-zero path. Groups 2/3 zero (<=2-D tensor).
static __device__ __forceinline__ void tdm_issue(unsigned lds_addr,
                                                 unsigned long long ga) {
    // Force the wave-uniform descriptor dwords into SGPR-friendly form.
    unsigned lds = (unsigned)__builtin_amdgcn_readfirstlane((int)lds_addr);
    unsigned glo = (unsigned)__builtin_amdgcn_readfirstlane((int)(unsigned)ga);
    unsigned ghi = (unsigned)__builtin_amdgcn_readfirstlane((int)(unsigned)(ga >> 32));

    v4u g0;
    g0.x = 1u;                                      // count=1 (valid descriptor)
    g0.y = lds;                                     // lds_addr (bytes)
    g0.z = glo;                                     // global_addr[31:0]
    g0.w = (ghi & 0x01FFFFFFu) | 0x80000000u;       // global_addr[56:32] | type=2

    v8i g1;
    g1[0] = 0x00020000;     // wg_mask=0, data_size=2 (4B), no pad/iter/barrier
    g1[1] = 0;              // atomic_barrier_addr=0, tensor_dim0 lo16 (2^30 -> 0)
    g1[2] = 0x4000;         // tensor_dim0 hi16 = 0x4000, tensor_dim1 lo16 = 0
    g1[3] = 0x00204000;     // tensor_dim1 hi16 = 0x4000, tile_dim0 = 32
    g1[4] = CH_T;           // tile_dim1 = 32 rows, tile_dim2 = 0
    g1[5] = DM_C;           // tensor_dim0_stride = 256 elements (lo32)
    g1[6] = 0;              // stride0 hi16 = 0, stride1 lo16 = 0
    g1[7] = 0;              // stride1 hi32 = 0

    v4i z4 = {0, 0, 0, 0};
#if defined(__clang_major__) && (__clang_major__ >= 23)
    v8i z8 = {0, 0, 0, 0, 0, 0, 0, 0};
    __builtin_amdgcn_tensor_load_to_lds(g0, g1, z4, z4, z8, 0);
#else
    __builtin_amdgcn_tensor_load_to_lds(g0, g1, z4, z4, 0);
#endif
}

__global__ __launch_bounds__(DM_CPB)
void DeltaModulator1Bit_kernel(const float* __restrict__ x,
                               const float* __restrict__ step,
                               float* __restrict__ bits,
                               float* __restrict__ y) {
    __shared__ float ring[NBUF][CH_T][DM_CPB];      // 64 KB LDS ring

    const int ctiles = DM_C / DM_CPB;               // 8 channel tiles per batch
    const int b    = blockIdx.x / ctiles;
    const int ct   = blockIdx.x - b * ctiles;
    const int c0   = ct * DM_CPB;
    const int lane = threadIdx.x;

    const float stepv = step[c0 + lane];            // step is [1, C]

    const size_t chunkElems = (size_t)CH_T * DM_C;  // elements per tile step
    const unsigned long long gbase =
        (unsigned long long)(uintptr_t)(x + ((size_t)b * DM_T) * DM_C + (size_t)c0);

    // ---- prologue: fill the TDM pipeline (AHEAD tiles in flight) ----
    for (int j = 0; j < AHEAD; ++j) {
        unsigned lds = (unsigned)(uintptr_t)&ring[j & (NBUF - 1)][0][0];
        tdm_issue(lds, gbase + (unsigned long long)j * (chunkElems * 4ull));
    }

    const size_t outbase = ((size_t)b * DM_T) * DM_C + (size_t)c0 + (size_t)lane;
    float* __restrict__ pb = bits + outbase;
    float* __restrict__ py = y + outbase;

    float prev = 0.0f;

    // ---- steady state: wait for oldest tile, consume it, reissue ----
    for (int i = 0; i < NCHUNK - AHEAD; ++i) {
        __builtin_amdgcn_s_wait_tensorcnt(AHEAD - 1);   // tile i landed in LDS
        const float* __restrict__ ls = &ring[i & (NBUF - 1)][0][lane];
#pragma unroll
        for (int tt = 0; tt < CH_T; ++tt) {
            const float xv  = ls[tt * DM_CPB];                   // ds_load_b32
            const bool  ge  = (xv >= prev);                      // diff >= 0
            const float sgn = ge ? 1.0f : -1.0f;
            const float cur = __builtin_fmaf(stepv, sgn, prev);  // exact: step*(+/-1)
            const float bit = ge ? 0.0f : 1.0f;
            const size_t o  = (size_t)tt * DM_C;
            __builtin_nontemporal_store(bit, pb + o);
            __builtin_nontemporal_store(cur, py + o);
            prev = cur;
        }
        pb += chunkElems;
        py += chunkElems;
        // Reissue into slot (i+AHEAD)&15 == (i-1)&15: consumed last iteration,
        // and DS ops are in-order per wave, so its reads have completed.
        const int nj = i + AHEAD;
        unsigned lds = (unsigned)(uintptr_t)&ring[nj & (NBUF - 1)][0][0];
        tdm_issue(lds, gbase + (unsigned long long)nj * (chunkElems * 4ull));
    }

    // ---- tail: no more issues; wait immediate must be constant -> drain ----
    for (int i = NCHUNK - AHEAD; i < NCHUNK; ++i) {
        __builtin_amdgcn_s_wait_tensorcnt(0);           // all remaining tiles done
        const float* __restrict__ ls = &ring[i & (NBUF - 1)][0][lane];
#pragma unroll
        for (int tt = 0; tt < CH_T; ++tt) {
            const float xv  = ls[tt * DM_CPB];
            const bool  ge  = (xv >= prev);
            const float sgn = ge ? 1.0f : -1.0f;
            const float cur = __builtin_fmaf(stepv, sgn, prev);
            const float bit = ge ? 0.0f : 1.0f;
            const size_t o  = (size_t)tt * DM_C;
            __builtin_nontemporal_store(bit, pb + o);
            __builtin_nontemporal_store(cur, py + o);
            prev = cur;
        }
        pb += chunkElems;
        py += chunkElems;
    }
}

extern "C" void kernel_launch(void* const* d_in, const int* in_sizes, int n_in,
                              void* d_out, int out_size, void* d_ws, size_t ws_size,
                              hipStream_t stream) {
    (void)in_sizes; (void)n_in; (void)out_size; (void)d_ws; (void)ws_size;

    const float* x    = (const float*)d_in[0];   // [B, T, C] fp32
    const float* step = (const float*)d_in[1];   // [1, C]    fp32

    float* bits = (float*)d_out;                                  // [B, T, C]
    float* y    = (float*)d_out + (size_t)DM_B * DM_T * DM_C;     // [B, T, C]

    dim3 grid(DM_B * (DM_C / DM_CPB));           // 128 blocks (1 wave each)
    dim3 block(DM_CPB);                          // 32 threads = 1 wave32

    DeltaModulator1Bit_kernel<<<grid, block, 0, stream>>>(x, step, bits, y);
}